// GCN_7859790152293
// MI455X (gfx1250) — compile-verified
//
#include <hip/hip_runtime.h>
#include <hip/hip_bf16.h>

typedef __attribute__((ext_vector_type(2))) float v2f;
typedef __attribute__((ext_vector_type(8))) float v8f;

#define BN_EPS 1e-5f

__device__ __forceinline__ void atomAddF(float* p, float v) {
  // lowers to global_atomic_add_f32 (L2-side f32 atomic) on gfx1250
  unsafeAtomicAdd(p, v);
}

// ---------------------------------------------------------------------------
// degree / normalization helpers
// ---------------------------------------------------------------------------
__global__ void fill_value_kernel(float* __restrict__ p, float v, int n) {
  int i = blockIdx.x * blockDim.x + threadIdx.x;
  if (i < n) p[i] = v;
}

__global__ void count_deg_kernel(const int* __restrict__ dst, float* __restrict__ deg, int E) {
  int i = blockIdx.x * blockDim.x + threadIdx.x;
  if (i < E) atomAddF(&deg[dst[i]], 1.0f);
}

__global__ void rsqrt_inplace_kernel(float* __restrict__ p, int n) {
  int i = blockIdx.x * blockDim.x + threadIdx.x;
  if (i < n) p[i] = rsqrtf(p[i]);
}

// ---------------------------------------------------------------------------
// Wt[col*K + k] = (col < dout) ? W[k*dout + col] : 0    (transpose + pad cols)
// ---------------------------------------------------------------------------
__global__ void transpose_pad_kernel(const float* __restrict__ W, float* __restrict__ Wt,
                                     int K, int dout, int dpad) {
  int i = blockIdx.x * blockDim.x + threadIdx.x;
  if (i >= dpad * K) return;
  int col = i / K;
  int k   = i - col * K;
  Wt[i] = (col < dout) ? W[k * dout + col] : 0.0f;
}

// ---------------------------------------------------------------------------
// f32 WMMA GEMM:  C[n x dpad] = A[n x K] @ Wt^T   (Wt is col-major, padded)
// One 16x16 tile per wave, V_WMMA_F32_16X16X4_F32, K stepped by 4.
// A 16x4 lane layout: lanes 0-15 hold K=k,k+1 ; lanes 16-31 hold K=k+2,k+3.
// B mirrors A via the col-major Wt. C/D: VGPR r -> row tileM*16+r (+8 for
// lanes 16-31), col = lane&15. dpad is a multiple of 16 -> no guards at all.
// ---------------------------------------------------------------------------
__global__ void gemm_f32_wmma_kernel(const float* __restrict__ A, const float* __restrict__ Wt,
                                     float* __restrict__ C, int K, int dpad, int tilesM) {
  int wave  = threadIdx.x >> 5;
  int tileM = blockIdx.x * 4 + wave;      // 4 waves / 128-thread block
  int tileN = blockIdx.y;
  if (tileM >= tilesM) return;            // wave-uniform: EXEC stays all-1s

  int lane = threadIdx.x & 31;
  int half = lane >> 4;                   // 0 = lanes 0-15, 1 = lanes 16-31
  int l15  = lane & 15;
  const float* aPtr = A  + (tileM * 16 + l15) * K + half * 2;
  const float* bPtr = Wt + (tileN * 16 + l15) * K + half * 2;

  v8f acc = {};
#pragma unroll 8
  for (int k = 0; k < K; k += 4) {
    v2f a = *(const v2f*)(aPtr + k);      // global_load_b64, 8B aligned
    v2f b = *(const v2f*)(bPtr + k);      // global_load_b64, 8B aligned
    acc = __builtin_amdgcn_wmma_f32_16x16x4_f32(false, a, false, b, (short)0, acc,
                                                false, false);
  }

  int colB  = tileN * 16 + l15;
  int rbase = tileM * 16 + half * 8;
#pragma unroll
  for (int r = 0; r < 8; ++r)
    C[(rbase + r) * dpad + colB] = acc[r];
}

// ---------------------------------------------------------------------------
// out[i][c] = bias[c] + dinv[i]^2 * h[i][c]   (bias + self-loop term),
// and zero the 256-entry BN stats buffer.
// ---------------------------------------------------------------------------
__global__ void init_out_kernel(float* __restrict__ out, const float* __restrict__ bias,
                                const float* __restrict__ h, const float* __restrict__ dinv,
                                int n, int d, int strideH, float* __restrict__ stats) {
  int gid = blockIdx.x * blockDim.x + threadIdx.x;
  if (gid < 256) stats[gid] = 0.0f;
  int total = n * d;
  for (int i = gid; i < total; i += gridDim.x * blockDim.x) {
    int row = i / d;
    int col = i - row * d;
    float di = dinv[row];
    out[i] = bias[col] + di * di * h[row * strideH + col];
  }
}

// ---------------------------------------------------------------------------
// edge aggregation (d == 128): one wave per edge, float4 per lane.
// out[dst] += dinv[src]*dinv[dst] * h[src]
// ---------------------------------------------------------------------------
__global__ void aggregate128_kernel(const float* __restrict__ h, const int* __restrict__ src,
                                    const int* __restrict__ dst, const float* __restrict__ dinv,
                                    float* __restrict__ out, int E) {
  int item = blockIdx.x * (blockDim.x >> 5) + (threadIdx.x >> 5);
  if (item >= E) return;
  int lane = threadIdx.x & 31;
  int s = src[item];
  int t = dst[item];
  float norm = dinv[s] * dinv[t];
  float4 v = *(const float4*)(h + s * 128 + lane * 4);   // global_load_b128
  float* o = out + t * 128 + lane * 4;
  atomAddF(o + 0, norm * v.x);
  atomAddF(o + 1, norm * v.y);
  atomAddF(o + 2, norm * v.z);
  atomAddF(o + 3, norm * v.w);
}

// generic strided variant (final layer: read stride 48, write d = 40)
__global__ void aggregate_strided_kernel(const float* __restrict__ h, const int* __restrict__ src,
                                         const int* __restrict__ dst, const float* __restrict__ dinv,
                                         float* __restrict__ out, int E, int strideH, int d) {
  int item = blockIdx.x * (blockDim.x >> 5) + (threadIdx.x >> 5);
  if (item >= E) return;
  int lane = threadIdx.x & 31;
  int s = src[item];
  int t = dst[item];
  float norm = dinv[s] * dinv[t];
  for (int f = lane; f < d; f += 32)
    atomAddF(&out[t * d + f], norm * h[s * strideH + f]);
}

// ---------------------------------------------------------------------------
// BN column stats: blockDim.x == d (128); register accumulation over 128 rows,
// then one atomic per column per block. stats[0..d-1]=sum, [d..2d-1]=sumsq.
// ---------------------------------------------------------------------------
__global__ void bn_stats_kernel(const float* __restrict__ h, float* __restrict__ stats,
                                int n, int d) {
  int col  = threadIdx.x;
  int row0 = blockIdx.x * 128;
  float s = 0.0f, s2 = 0.0f;
  for (int r = 0; r < 128; ++r) {
    int row = row0 + r;
    if (row < n) {
      float v = h[row * d + col];
      s += v; s2 += v * v;
    }
  }
  atomAddF(&stats[col], s);
  atomAddF(&stats[d + col], s2);
}

__global__ void bn_apply_relu_kernel(float* __restrict__ h, const float* __restrict__ stats,
                                     const float* __restrict__ g, const float* __restrict__ bt,
                                     int n, int d) {
  int total = n * d;
  float inv_n = 1.0f / (float)n;
  for (int i = blockIdx.x * blockDim.x + threadIdx.x; i < total;
       i += gridDim.x * blockDim.x) {
    int col  = i & (d - 1);                 // d == 128
    float mu  = stats[col] * inv_n;
    float var = stats[d + col] * inv_n - mu * mu;
    float v   = (h[i] - mu) * rsqrtf(var + BN_EPS) * g[col] + bt[col];
    h[i] = v > 0.0f ? v : 0.0f;
  }
}

// ---------------------------------------------------------------------------
// in-place row log-softmax, one wave32 per row (d <= 64)
// ---------------------------------------------------------------------------
__global__ void log_softmax_kernel(float* __restrict__ out, int n, int d) {
  int row = blockIdx.x * (blockDim.x >> 5) + (threadIdx.x >> 5);
  if (row >= n) return;
  int lane = threadIdx.x & 31;
  float ninf = -__builtin_inff();
  float v0 = (lane      < d) ? out[row * d + lane]      : ninf;
  float v1 = (lane + 32 < d) ? out[row * d + lane + 32] : ninf;
  float m = fmaxf(v0, v1);
#pragma unroll
  for (int off = 16; off > 0; off >>= 1) m = fmaxf(m, __shfl_xor(m, off, 32));
  float s = ((lane < d) ? expf(v0 - m) : 0.0f) + ((lane + 32 < d) ? expf(v1 - m) : 0.0f);
#pragma unroll
  for (int off = 16; off > 0; off >>= 1) s += __shfl_xor(s, off, 32);
  float lg = m + logf(s);
  if (lane      < d) out[row * d + lane]      = v0 - lg;
  if (lane + 32 < d) out[row * d + lane + 32] = v1 - lg;
}

// ---------------------------------------------------------------------------
extern "C" void kernel_launch(void* const* d_in, const int* in_sizes, int n_in,
                              void* d_out, int out_size, void* d_ws, size_t ws_size,
                              hipStream_t stream) {
  const float* x   = (const float*)d_in[0];
  const int*   ei  = (const int*)  d_in[1];
  const float* W0  = (const float*)d_in[2];
  const float* b0  = (const float*)d_in[3];
  const float* W1  = (const float*)d_in[4];
  const float* b1  = (const float*)d_in[5];
  const float* W2  = (const float*)d_in[6];
  const float* b2  = (const float*)d_in[7];
  const float* g0  = (const float*)d_in[8];
  const float* bt0 = (const float*)d_in[9];
  const float* g1  = (const float*)d_in[10];
  const float* bt1 = (const float*)d_in[11];

  const int D     = 128;                   // D_IN == D_H
  const int DOUT  = in_sizes[7];           // len(b2) == 40
  const int DPAD  = (DOUT + 15) & ~15;     // 48
  const int N     = in_sizes[0] / D;
  const int E     = in_sizes[1] / 2;
  const int* src  = ei;
  const int* dst  = ei + E;

  // workspace layout (floats)
  float* ws    = (float*)d_ws;
  float* dinv  = ws;                        // N      degree -> dinv in place
  float* bufA  = dinv + N;                  // N*D    transform output
  float* bufB  = bufA + (size_t)N * D;      // N*D    aggregation output
  float* bufO  = bufB + (size_t)N * D;      // N*DPAD final transform output
  float* Wt    = bufO + (size_t)N * DPAD;   // D*D    transposed/padded weights
  float* stats = Wt + (size_t)D * D;        // 256    BN sums / sumsq
  float* outF  = (float*)d_out;             // N*DOUT

  const int tilesM = (N + 15) / 16;
  dim3 gemmBlk(128);
  dim3 gemmGridH((tilesM + 3) / 4, D / 16);      // 8 column tiles
  dim3 gemmGridO((tilesM + 3) / 4, DPAD / 16);   // 3 column tiles
  int aggBlocks = (E + 7) / 8;                   // 8 waves / 256-thread block

  // --- degree / normalization (self-loop counts as 1) ---
  fill_value_kernel<<<(N + 255) / 256, 256, 0, stream>>>(dinv, 1.0f, N);
  count_deg_kernel<<<(E + 255) / 256, 256, 0, stream>>>(dst, dinv, E);
  rsqrt_inplace_kernel<<<(N + 255) / 256, 256, 0, stream>>>(dinv, N);

  // --- layer 0: x @ W0 -> (bias + self-loop) -> edge scatter -> BN -> ReLU ---
  transpose_pad_kernel<<<(D * D + 255) / 256, 256, 0, stream>>>(W0, Wt, D, D, D);
  gemm_f32_wmma_kernel<<<gemmGridH, gemmBlk, 0, stream>>>(x, Wt, bufA, D, D, tilesM);
  init_out_kernel<<<1024, 256, 0, stream>>>(bufB, b0, bufA, dinv, N, D, D, stats);
  aggregate128_kernel<<<aggBlocks, 256, 0, stream>>>(bufA, src, dst, dinv, bufB, E);
  bn_stats_kernel<<<(N + 127) / 128, D, 0, stream>>>(bufB, stats, N, D);
  bn_apply_relu_kernel<<<2048, 256, 0, stream>>>(bufB, stats, g0, bt0, N, D);

  // --- layer 1 ---
  transpose_pad_kernel<<<(D * D + 255) / 256, 256, 0, stream>>>(W1, Wt, D, D, D);
  gemm_f32_wmma_kernel<<<gemmGridH, gemmBlk, 0, stream>>>(bufB, Wt, bufA, D, D, tilesM);
  init_out_kernel<<<1024, 256, 0, stream>>>(bufB, b1, bufA, dinv, N, D, D, stats);
  aggregate128_kernel<<<aggBlocks, 256, 0, stream>>>(bufA, src, dst, dinv, bufB, E);
  bn_stats_kernel<<<(N + 127) / 128, D, 0, stream>>>(bufB, stats, N, D);
  bn_apply_relu_kernel<<<2048, 256, 0, stream>>>(bufB, stats, g1, bt1, N, D);

  // --- layer 2: transform -> aggregate into d_out -> log_softmax in place ---
  transpose_pad_kernel<<<(D * DPAD + 255) / 256, 256, 0, stream>>>(W2, Wt, D, DOUT, DPAD);
  gemm_f32_wmma_kernel<<<gemmGridO, gemmBlk, 0, stream>>>(bufB, Wt, bufO, D, DPAD, tilesM);
  init_out_kernel<<<1024, 256, 0, stream>>>(outF, b2, bufO, dinv, N, DOUT, DPAD, stats);
  aggregate_strided_kernel<<<aggBlocks, 256, 0, stream>>>(bufO, src, dst, dinv, outF, E, DPAD, DOUT);
  log_softmax_kernel<<<(N + 7) / 8, 256, 0, stream>>>(outF, N, DOUT);
}